// get_model_55645596287823
// MI455X (gfx1250) — compile-verified
//
#include <hip/hip_runtime.h>

#define NSAMP 32
#define NCLS  13

typedef __attribute__((ext_vector_type(2))) float v2f;
typedef __attribute__((ext_vector_type(8))) float v8f;

static inline unsigned cdivu(size_t a, unsigned b) { return (unsigned)((a + b - 1) / b); }

// =====================================================================
// GEMM via CDNA5 fp32 WMMA: C[M,N] = A[M,K] @ B + bias (+ReLU), where the
// weight matrix is supplied TRANSPOSED: WT[N,K] (row stride K).  This makes
// both A and B per-lane fragment loads contiguous in K, so the compiler can
// emit single global_load_b64 per fragment with immediate offsets (verified
// for the A side in round 1 asm).
// One wave per 16x32 output strip (two 16x16 tiles sharing the A fragment).
// A frag (ISA 7.12.2): lanes0-15 M=0..15 K{0,1}; lanes16-31 M=0..15 K{2,3}
// B frag mirrored with lane = N.  C/D: VGPR r -> M = r + 8*(lane/16), N = lane%16.
// M/N edges: indices CLAMPED (always-in-bounds loads, no exec masking);
// out-of-range rows/cols only feed D elements the guarded store never writes.
// K tail is the only case needing zero-masking (it would pollute every output).
// =====================================================================
__global__ void __launch_bounds__(32)
gemm_wmma(const float* __restrict__ A, const float* __restrict__ WT,
          const float* __restrict__ bias, float* __restrict__ C,
          int M, int K, int N, int relu)
{
  const int lane = threadIdx.x;
  const int half = lane >> 4;
  const int l16  = lane & 15;
  const int tm = blockIdx.y << 4;
  const int tn = blockIdx.x << 5;          // 32 output columns per wave
  const int row   = tm + l16;
  const int rowc  = (row  < M) ? row  : (M - 1);
  const int col0  = tn + l16;
  const int col1  = tn + 16 + l16;
  const int col0c = (col0 < N) ? col0 : (N - 1);
  const int col1c = (col1 < N) ? col1 : (N - 1);

  const float* __restrict__ Ar  = A  + (size_t)rowc  * K + (half << 1);
  const float* __restrict__ Bt0 = WT + (size_t)col0c * K + (half << 1);
  const float* __restrict__ Bt1 = WT + (size_t)col1c * K + (half << 1);

  v8f acc0 = {0.f, 0.f, 0.f, 0.f, 0.f, 0.f, 0.f, 0.f};
  v8f acc1 = {0.f, 0.f, 0.f, 0.f, 0.f, 0.f, 0.f, 0.f};

  const int K4 = K & ~3;
  int k = 0;
#pragma unroll 4
  for (; k < K4; k += 4) {
    v2f a, b0, b1;
    a.x  = Ar[k];
    a.y  = Ar[k + 1];
    b0.x = Bt0[k];
    b0.y = Bt0[k + 1];
    b1.x = Bt1[k];
    b1.y = Bt1[k + 1];
#if __has_builtin(__builtin_amdgcn_wmma_f32_16x16x4_f32)
    acc0 = __builtin_amdgcn_wmma_f32_16x16x4_f32(false, a, false, b0,
                                                 (short)0, acc0, false, false);
    acc1 = __builtin_amdgcn_wmma_f32_16x16x4_f32(false, a, false, b1,
                                                 (short)0, acc1, false, false);
#else
    for (int r = 0; r < 8; ++r) { acc0[r] += a.x * b0.x; acc1[r] += a.x * b1.x; }
#endif
  }
  if (k < K) {                              // K % 4 tail: branch-free clamp+mask
    const int ka  = k + (half << 1);
    const int k0  = (ka     < K) ? ka     : (K - 1);
    const int k1  = (ka + 1 < K) ? ka + 1 : (K - 1);
    const float m0 = (ka     < K) ? 1.f : 0.f;
    const float m1 = (ka + 1 < K) ? 1.f : 0.f;
    v2f a, b0, b1;
    a.x  = m0 * A[(size_t)rowc * K + k0];
    a.y  = m1 * A[(size_t)rowc * K + k1];
    b0.x = m0 * WT[(size_t)col0c * K + k0];
    b0.y = m1 * WT[(size_t)col0c * K + k1];
    b1.x = m0 * WT[(size_t)col1c * K + k0];
    b1.y = m1 * WT[(size_t)col1c * K + k1];
#if __has_builtin(__builtin_amdgcn_wmma_f32_16x16x4_f32)
    acc0 = __builtin_amdgcn_wmma_f32_16x16x4_f32(false, a, false, b0,
                                                 (short)0, acc0, false, false);
    acc1 = __builtin_amdgcn_wmma_f32_16x16x4_f32(false, a, false, b1,
                                                 (short)0, acc1, false, false);
#else
    for (int r = 0; r < 8; ++r) { acc0[r] += a.x * b0.x; acc1[r] += a.x * b1.x; }
#endif
  }

  if (col0 < N) {
    const float bv = bias ? bias[col0] : 0.f;
#pragma unroll
    for (int r = 0; r < 8; ++r) {
      const int rr = tm + r + (half << 3);
      if (rr < M) {
        float v = acc0[r] + bv;
        if (relu) v = fmaxf(v, 0.f);
        C[(size_t)rr * N + col0] = v;
      }
    }
  }
  if (col1 < N) {
    const float bv = bias ? bias[col1] : 0.f;
#pragma unroll
    for (int r = 0; r < 8; ++r) {
      const int rr = tm + r + (half << 3);
      if (rr < M) {
        float v = acc1[r] + bv;
        if (relu) v = fmaxf(v, 0.f);
        C[(size_t)rr * N + col1] = v;
      }
    }
  }
}

// ------------------------------------------- weight transpose W[K,N]->WT[N,K]
__global__ void k_transpose(const float* __restrict__ W, float* __restrict__ WT,
                            size_t total, int K, int N)
{
  size_t t = (size_t)blockIdx.x * blockDim.x + threadIdx.x;
  if (t >= total) return;
  const int n = (int)(t % N);
  const int k = (int)(t / N);
  WT[(size_t)n * K + k] = W[t];
}

// ---------------------------------------------------------- concat [a|b] rows
__global__ void k_concat_rows(const float* __restrict__ a, const float* __restrict__ b,
                              float* __restrict__ y, size_t total, int C1, int C2)
{
  size_t t = (size_t)blockIdx.x * blockDim.x + threadIdx.x;
  if (t >= total) return;
  const int C = C1 + C2;
  const int c = (int)(t % C);
  const size_t i = t / C;
  y[t] = (c < C1) ? a[i * C1 + c] : b[i * C2 + (c - C1)];
}

// ------------------------------------------------------- furthest point sample
__global__ void __launch_bounds__(1024)
k_fps(const float* __restrict__ p, int n, int m,
      int* __restrict__ idx, float* __restrict__ dist)
{
  __shared__ float sv[1024];
  __shared__ int   si[1024];
  __shared__ int   slast;
  const int t = threadIdx.x;
  for (int i = t; i < n; i += 1024) dist[i] = 1e10f;
  if (t == 0) { idx[0] = 0; slast = 0; }
  __syncthreads();
  for (int s = 1; s < m; ++s) {
    const int last = slast;
    const float lx = p[3 * last], ly = p[3 * last + 1], lz = p[3 * last + 2];
    float bv = -1.f; int bi = n - 1;
    for (int i = t; i < n; i += 1024) {
      const float dx = p[3 * i] - lx, dy = p[3 * i + 1] - ly, dz = p[3 * i + 2] - lz;
      const float d = dx * dx + dy * dy + dz * dz;
      float dm = dist[i];
      if (d < dm) { dm = d; dist[i] = dm; }
      if (dm > bv) { bv = dm; bi = i; }     // strict > keeps lowest index on ties
    }
    sv[t] = bv; si[t] = bi;
    __syncthreads();
    for (int off = 512; off > 0; off >>= 1) {
      if (t < off) {
        const float v2 = sv[t + off]; const int i2 = si[t + off];
        if (v2 > sv[t] || (v2 == sv[t] && i2 < si[t])) { sv[t] = v2; si[t] = i2; }
      }
      __syncthreads();
    }
    if (t == 0) { idx[s] = si[0]; slast = si[0]; }
    __syncthreads();
  }
}

// --------------------------------------------------- kNN (k<=32, top_k(-d) order)
__global__ void k_knn(const float* __restrict__ q, int nq,
                      const float* __restrict__ ref, int nref,
                      int k, int* __restrict__ out)
{
  const int i = blockIdx.x * blockDim.x + threadIdx.x;
  if (i >= nq) return;
  const float qx = q[3 * i], qy = q[3 * i + 1], qz = q[3 * i + 2];
  const float qq = qx * qx + qy * qy + qz * qz;
  float bd[32]; int bi[32];
  for (int j = 0; j < 32; ++j) { bd[j] = 3.4e38f; bi[j] = 0; }
  for (int r = 0; r < nref; ++r) {
    const float rx = ref[3 * r], ry = ref[3 * r + 1], rz = ref[3 * r + 2];
    const float d = qq - 2.f * (qx * rx + qy * ry + qz * rz) + (rx * rx + ry * ry + rz * rz);
    if (d < bd[k - 1]) {
      int j = k - 1;
      while (j > 0 && bd[j - 1] > d) { bd[j] = bd[j - 1]; bi[j] = bi[j - 1]; --j; }
      bd[j] = d; bi[j] = r;
    }
  }
  for (int j = 0; j < k; ++j) out[(size_t)i * k + j] = bi[j];
}

// -------------------------------------------------------------- row gather
__global__ void k_gather_rows(const float* __restrict__ src, const int* __restrict__ idx,
                              float* __restrict__ dst, size_t total, int C)
{
  size_t t = (size_t)blockIdx.x * blockDim.x + threadIdx.x;
  if (t >= total) return;
  const int c = (int)(t % C);
  const size_t i = t / C;
  dst[t] = src[(size_t)idx[i] * C + c];
}

// -------------------------------- transition-down grouping: [p[gi]-np | x[gi]]
__global__ void k_group_td(const float* __restrict__ p, const float* __restrict__ x,
                           const float* __restrict__ np_, const int* __restrict__ gidx,
                           float* __restrict__ g, size_t total, int cin)
{
  size_t t = (size_t)blockIdx.x * blockDim.x + threadIdx.x;
  if (t >= total) return;
  const int C = 3 + cin;
  const int c = (int)(t % C);
  const size_t ij = t / C;
  const int j = (int)(ij % NSAMP);
  const size_t i = ij / NSAMP;
  const int src = gidx[i * NSAMP + j];
  g[t] = (c < 3) ? (p[3 * src + c] - np_[3 * i + c])
                 : x[(size_t)src * cin + (c - 3)];
}

// --------------------------------------- BN (train mode): per-channel stats
__global__ void k_bn_stats(const float* __restrict__ x, size_t M, int C,
                           float* __restrict__ mean, float* __restrict__ rstd)
{
  __shared__ float s1[256], s2[256];
  const int c = blockIdx.x;
  float a = 0.f, b = 0.f;
  for (size_t r = threadIdx.x; r < M; r += 256) {
    const float v = x[r * C + c];
    a += v; b += v * v;
  }
  s1[threadIdx.x] = a; s2[threadIdx.x] = b;
  __syncthreads();
  for (int off = 128; off > 0; off >>= 1) {
    if ((int)threadIdx.x < off) { s1[threadIdx.x] += s1[threadIdx.x + off];
                                  s2[threadIdx.x] += s2[threadIdx.x + off]; }
    __syncthreads();
  }
  if (threadIdx.x == 0) {
    const float mu = s1[0] / (float)M;
    const float var = s2[0] / (float)M - mu * mu;
    mean[c] = mu; rstd[c] = rsqrtf(var + 1e-5f);
  }
}

// out = relu( (x-mean)*rstd + (res ? res : 0) )
__global__ void k_bn_apply(const float* __restrict__ x, float* __restrict__ out,
                           size_t total, int C, const float* __restrict__ mean,
                           const float* __restrict__ rstd, const float* __restrict__ res)
{
  size_t t = (size_t)blockIdx.x * blockDim.x + threadIdx.x;
  if (t >= total) return;
  const int c = (int)(t % C);
  float v = (x[t] - mean[c]) * rstd[c];
  if (res) v += res[t];
  out[t] = fmaxf(v, 0.f);
}

// ----------------------------------------------------- max-pool over NS axis
__global__ void k_maxpool(const float* __restrict__ h, float* __restrict__ out,
                          size_t total, int C)
{
  size_t t = (size_t)blockIdx.x * blockDim.x + threadIdx.x;
  if (t >= total) return;
  const int c = (int)(t % C);
  const size_t i = t / C;
  float v = -3.4e38f;
  for (int j = 0; j < NSAMP; ++j)
    v = fmaxf(v, h[(i * NSAMP + j) * C + c]);
  out[t] = v;
}

// ------------------------------------------- pr = p[idx] - p (rel coordinates)
__global__ void k_rel(const float* __restrict__ p, const int* __restrict__ idx,
                      float* __restrict__ pr, size_t total)
{
  size_t t = (size_t)blockIdx.x * blockDim.x + threadIdx.x;
  if (t >= total) return;
  const int c = (int)(t % 3);
  const size_t ij = t / 3;
  const int j = (int)(ij % NSAMP);
  const size_t i = ij / NSAMP;
  pr[t] = p[3 * idx[i * NSAMP + j] + c] - p[3 * i + c];
}

// ---------- fused: wpre = xk[idx]-xq+pr2 ; gv = xv[idx]+pr2
__global__ void k_qkv(const float* __restrict__ xk, const float* __restrict__ xq,
                      const float* __restrict__ xv, const float* __restrict__ pr2,
                      const int* __restrict__ idx, float* __restrict__ wpre,
                      float* __restrict__ gv, size_t total, int C)
{
  size_t t = (size_t)blockIdx.x * blockDim.x + threadIdx.x;
  if (t >= total) return;
  const int c = (int)(t % C);
  const size_t ij = t / C;
  const int j = (int)(ij % NSAMP);
  const size_t i = ij / NSAMP;
  const int s = idx[i * NSAMP + j];
  const float pv = pr2[t];
  wpre[t] = xk[(size_t)s * C + c] - xq[i * C + c] + pv;
  gv[t]   = xv[(size_t)s * C + c] + pv;
}

// -------------------------------------- softmax across NS for each (row, chan)
__global__ void k_softmax_ns(float* __restrict__ w, size_t total, int Cw)
{
  size_t t = (size_t)blockIdx.x * blockDim.x + threadIdx.x;
  if (t >= total) return;
  const int b = (int)(t % Cw);
  const size_t i = t / Cw;
  float mx = -3.4e38f;
  for (int j = 0; j < NSAMP; ++j) mx = fmaxf(mx, w[(i * NSAMP + j) * Cw + b]);
  float s = 0.f;
  for (int j = 0; j < NSAMP; ++j) s += __expf(w[(i * NSAMP + j) * Cw + b] - mx);
  const float inv = 1.f / s;
  for (int j = 0; j < NSAMP; ++j) {
    const size_t o = (i * NSAMP + j) * Cw + b;
    w[o] = __expf(w[o] - mx) * inv;
  }
}

// ------------- out[i,c] = sum_j gv[i,j,c] * w[i,j, c % Cw]
__global__ void k_attn(const float* __restrict__ gv, const float* __restrict__ w,
                       float* __restrict__ out, size_t total, int C, int Cw)
{
  size_t t = (size_t)blockIdx.x * blockDim.x + threadIdx.x;
  if (t >= total) return;
  const int c = (int)(t % C);
  const size_t i = t / C;
  const int b = c % Cw;
  float s = 0.f;
  for (int j = 0; j < NSAMP; ++j)
    s += gv[(i * NSAMP + j) * C + c] * w[(i * NSAMP + j) * Cw + b];
  out[t] = s;
}

// --------------------------------------------------------------- column mean
__global__ void k_colmean(const float* __restrict__ x, float* __restrict__ out,
                          int M, int C)
{
  const int c = blockIdx.x * blockDim.x + threadIdx.x;
  if (c >= C) return;
  float s = 0.f;
  for (int r = 0; r < M; ++r) s += x[(size_t)r * C + c];
  out[c] = s / (float)M;
}

// --------------------------------------------- concat row-wise with broadcast
__global__ void k_concat_bcast(const float* __restrict__ x, const float* __restrict__ g,
                               float* __restrict__ y, size_t total, int C1, int C2)
{
  size_t t = (size_t)blockIdx.x * blockDim.x + threadIdx.x;
  if (t >= total) return;
  const int C = C1 + C2;
  const int c = (int)(t % C);
  const size_t i = t / C;
  y[t] = (c < C1) ? x[i * C1 + c] : g[c - C1];
}

// --------------------------------- 3-NN inverse-distance interpolation weights
__global__ void k_interp_w(const float* __restrict__ pd, const float* __restrict__ ps,
                           const int* __restrict__ idx, float* __restrict__ w, int m)
{
  const int i = blockIdx.x * blockDim.x + threadIdx.x;
  if (i >= m) return;
  float ws[3]; float s = 0.f;
  for (int kk = 0; kk < 3; ++kk) {
    const int j = idx[(size_t)i * 3 + kk];
    const float dx = pd[3 * i] - ps[3 * j];
    const float dy = pd[3 * i + 1] - ps[3 * j + 1];
    const float dz = pd[3 * i + 2] - ps[3 * j + 2];
    const float d = sqrtf(dx * dx + dy * dy + dz * dz);
    ws[kk] = 1.f / (d + 1e-8f);
    s += ws[kk];
  }
  for (int kk = 0; kk < 3; ++kk) w[(size_t)i * 3 + kk] = ws[kk] / s;
}

__global__ void k_interp_apply(const float* __restrict__ feat, const int* __restrict__ idx,
                               const float* __restrict__ w, const float* __restrict__ addto,
                               float* __restrict__ out, size_t total, int C)
{
  size_t t = (size_t)blockIdx.x * blockDim.x + threadIdx.x;
  if (t >= total) return;
  const int c = (int)(t % C);
  const size_t i = t / C;
  float s = 0.f;
  for (int kk = 0; kk < 3; ++kk)
    s += feat[(size_t)idx[i * 3 + kk] * C + c] * w[i * 3 + kk];
  out[t] = s + (addto ? addto[t] : 0.f);
}

// --------------------------------------------------------------- log-softmax
__global__ void k_log_softmax(const float* __restrict__ x, float* __restrict__ out,
                              int n, int C)
{
  const int i = blockIdx.x * blockDim.x + threadIdx.x;
  if (i >= n) return;
  const float* r = x + (size_t)i * C;
  float mx = -3.4e38f;
  for (int c = 0; c < C; ++c) mx = fmaxf(mx, r[c]);
  float s = 0.f;
  for (int c = 0; c < C; ++c) s += __expf(r[c] - mx);
  const float l = __logf(s);
  for (int c = 0; c < C; ++c) out[(size_t)i * C + c] = r[c] - mx - l;
}

// =====================================================================
// Host-side orchestration
// =====================================================================
namespace {

struct Pool {
  char* base; size_t off;
  float* f(size_t n) { float* p = (float*)(base + off); off += ((n * 4 + 255) / 256) * 256; return p; }
  int*   i(size_t n) { int*   p = (int*)  (base + off); off += ((n * 4 + 255) / 256) * 256; return p; }
};

struct TrW  { const float *wq,*bq,*wk,*bk,*wv,*bv,*p1w,*p1b,*p2w,*p2b,*w1w,*w1b,*w2w,*w2b; };
struct BlkW { const float* l1; TrW tr; const float* l3; };
struct EncW { const float* td; BlkW b1, b2; };
struct DecW { const float *l1w,*l1b,*l2w,*l2b; };

// Transposes W into pool scratch, then runs the WMMA GEMM.  Pool is taken by
// value; later allocations in the caller may alias WT, which is safe because
// all kernels are stream-ordered (WT is fully consumed before any later write).
inline void gemm(hipStream_t st, Pool pool, const float* A, const float* W,
                 const float* bias, float* C, int M, int K, int N, int relu) {
  float* WT = pool.f((size_t)K * N);
  k_transpose<<<cdivu((size_t)K * N, 256), 256, 0, st>>>(W, WT, (size_t)K * N, K, N);
  dim3 g((unsigned)((N + 31) / 32), (unsigned)((M + 15) / 16), 1);
  gemm_wmma<<<g, 32, 0, st>>>(A, WT, bias, C, M, K, N, relu);
}

inline void bn_act(hipStream_t st, const float* x, float* out, size_t M, int C,
                   const float* res, float* mean, float* rstd) {
  k_bn_stats<<<(unsigned)C, 256, 0, st>>>(x, M, C, mean, rstd);
  const size_t tot = M * (size_t)C;
  k_bn_apply<<<cdivu(tot, 256), 256, 0, st>>>(x, out, tot, C, mean, rstd, res);
}

void pt_layer_run(hipStream_t st, const float* p, const float* x, const TrW& W,
                  int n, int C, const int* idx, float* out, Pool pool,
                  float* mean, float* rstd) {
  const int Cw = C / 8;
  const size_t nNS = (size_t)n * NSAMP;
  float* xq = pool.f((size_t)n * C);
  float* xk = pool.f((size_t)n * C);
  float* xv = pool.f((size_t)n * C);
  gemm(st, pool, x, W.wq, W.bq, xq, n, C, C, 0);
  gemm(st, pool, x, W.wk, W.bk, xk, n, C, C, 0);
  gemm(st, pool, x, W.wv, W.bv, xv, n, C, C, 0);
  float* prr = pool.f(nNS * 3);
  k_rel<<<cdivu(nNS * 3, 256), 256, 0, st>>>(p, idx, prr, nNS * 3);
  float* pra = pool.f(nNS * 3);
  gemm(st, pool, prr, W.p1w, W.p1b, pra, (int)nNS, 3, 3, 0);
  float* prb = pool.f(nNS * 3);
  bn_act(st, pra, prb, nNS, 3, nullptr, mean, rstd);
  float* pr2 = pool.f(nNS * C);
  gemm(st, pool, prb, W.p2w, W.p2b, pr2, (int)nNS, 3, C, 0);
  float* wpre = pool.f(nNS * C);
  float* gv   = pool.f(nNS * C);
  k_qkv<<<cdivu(nNS * C, 256), 256, 0, st>>>(xk, xq, xv, pr2, idx, wpre, gv, nNS * C, C);
  float* wa = pool.f(nNS * C);
  bn_act(st, wpre, wa, nNS, C, nullptr, mean, rstd);
  float* w1 = pool.f(nNS * Cw);
  gemm(st, pool, wa, W.w1w, W.w1b, w1, (int)nNS, C, Cw, 0);
  float* w1r = pool.f(nNS * Cw);
  bn_act(st, w1, w1r, nNS, Cw, nullptr, mean, rstd);
  float* w2 = pool.f(nNS * Cw);
  gemm(st, pool, w1r, W.w2w, W.w2b, w2, (int)nNS, Cw, Cw, 0);
  k_softmax_ns<<<cdivu((size_t)n * Cw, 256), 256, 0, st>>>(w2, (size_t)n * Cw, Cw);
  k_attn<<<cdivu((size_t)n * C, 256), 256, 0, st>>>(gv, w2, out, (size_t)n * C, C, Cw);
}

void pt_block_run(hipStream_t st, const float* p, const float* x, const BlkW& W,
                  int n, int C, const int* idx, float* out, Pool pool,
                  float* mean, float* rstd) {
  float* y0 = pool.f((size_t)n * C);
  gemm(st, pool, x, W.l1, nullptr, y0, n, C, C, 0);
  float* y1 = pool.f((size_t)n * C);
  bn_act(st, y0, y1, n, C, nullptr, mean, rstd);
  float* pt = pool.f((size_t)n * C);
  pt_layer_run(st, p, y1, W.tr, n, C, idx, pt, pool, mean, rstd);
  float* y2 = pool.f((size_t)n * C);
  bn_act(st, pt, y2, n, C, nullptr, mean, rstd);
  float* y3 = pool.f((size_t)n * C);
  gemm(st, pool, y2, W.l3, nullptr, y3, n, C, C, 0);
  bn_act(st, y3, out, n, C, /*residual pre-ReLU*/ x, mean, rstd);
}

void encoder_run(hipStream_t st, const float* p_in, const float* x_in, int n, int cin,
                 const EncW& W, int cout, float* p_out, float* x_out,
                 Pool pool, float* mean, float* rstd) {
  const int m = n / 4;
  int* fidx = pool.i((size_t)m);
  float* dist = pool.f((size_t)n);
  k_fps<<<1, 1024, 0, st>>>(p_in, n, m, fidx, dist);
  k_gather_rows<<<cdivu((size_t)m * 3, 256), 256, 0, st>>>(p_in, fidx, p_out, (size_t)m * 3, 3);
  int* gidx = pool.i((size_t)m * NSAMP);
  k_knn<<<cdivu((size_t)m, 128), 128, 0, st>>>(p_out, m, p_in, n, NSAMP, gidx);
  const int cg = 3 + cin;
  float* g = pool.f((size_t)m * NSAMP * cg);
  k_group_td<<<cdivu((size_t)m * NSAMP * cg, 256), 256, 0, st>>>(p_in, x_in, p_out, gidx, g,
                                                                 (size_t)m * NSAMP * cg, cin);
  float* h = pool.f((size_t)m * NSAMP * cout);
  gemm(st, pool, g, W.td, nullptr, h, m * NSAMP, cg, cout, 0);
  float* h2 = pool.f((size_t)m * NSAMP * cout);
  bn_act(st, h, h2, (size_t)m * NSAMP, cout, nullptr, mean, rstd);
  float* xa = pool.f((size_t)m * cout);
  k_maxpool<<<cdivu((size_t)m * cout, 256), 256, 0, st>>>(h2, xa, (size_t)m * cout, cout);
  int* pidx = pool.i((size_t)m * NSAMP);
  k_knn<<<cdivu((size_t)m, 128), 128, 0, st>>>(p_out, m, p_out, m, NSAMP, pidx);
  float* xb = pool.f((size_t)m * cout);
  pt_block_run(st, p_out, xa, W.b1, m, cout, pidx, xb, pool, mean, rstd);
  pt_block_run(st, p_out, xb, W.b2, m, cout, pidx, x_out, pool, mean, rstd);
}

void tu_head_run(hipStream_t st, const float* x, int n, int C, const DecW& W,
                 float* out, Pool pool, float* mean, float* rstd) {
  float* mu = pool.f((size_t)C);
  k_colmean<<<cdivu((size_t)C, 128), 128, 0, st>>>(x, mu, n, C);
  float* g = pool.f((size_t)C);
  gemm(st, pool, mu, W.l2w, W.l2b, g, 1, C, C, 1);  // ReLU fused
  float* y = pool.f((size_t)n * 2 * C);
  k_concat_bcast<<<cdivu((size_t)n * 2 * C, 256), 256, 0, st>>>(x, g, y, (size_t)n * 2 * C, C, C);
  float* y2 = pool.f((size_t)n * C);
  gemm(st, pool, y, W.l1w, W.l1b, y2, n, 2 * C, C, 0);
  bn_act(st, y2, out, n, C, nullptr, mean, rstd);
}

void tu_run(hipStream_t st, const float* p_f, const float* x_f, int n_f, int c_f,
            const float* p_c, const float* x_c, int n_c, int c_c,
            const DecW& W, float* out, Pool pool, float* mean, float* rstd) {
  float* f0 = pool.f((size_t)n_c * c_f);
  gemm(st, pool, x_c, W.l2w, W.l2b, f0, n_c, c_c, c_f, 0);
  float* f2 = pool.f((size_t)n_c * c_f);
  bn_act(st, f0, f2, n_c, c_f, nullptr, mean, rstd);
  int* id3 = pool.i((size_t)n_f * 3);
  k_knn<<<cdivu((size_t)n_f, 128), 128, 0, st>>>(p_f, n_f, p_c, n_c, 3, id3);
  float* w3 = pool.f((size_t)n_f * 3);
  k_interp_w<<<cdivu((size_t)n_f, 256), 256, 0, st>>>(p_f, p_c, id3, w3, n_f);
  const float* addto = nullptr;
  if (x_f) {
    float* b0 = pool.f((size_t)n_f * c_f);
    gemm(st, pool, x_f, W.l1w, W.l1b, b0, n_f, c_f, c_f, 0);
    float* b1 = pool.f((size_t)n_f * c_f);
    bn_act(st, b0, b1, n_f, c_f, nullptr, mean, rstd);
    addto = b1;
  }
  k_interp_apply<<<cdivu((size_t)n_f * c_f, 256), 256, 0, st>>>(f2, id3, w3, addto, out,
                                                                (size_t)n_f * c_f, c_f);
}

} // namespace

extern "C" void kernel_launch(void* const* d_in, const int* in_sizes, int n_in,
                              void* d_out, int out_size, void* d_ws, size_t ws_size,
                              hipStream_t stream) {
  (void)n_in; (void)out_size; (void)ws_size;
  const float* p0 = (const float*)d_in[0];
  const float* x0 = (const float*)d_in[1];
  const int N = in_sizes[0] / 3;                 // 16384
  const int n1 = N / 4, n2 = N / 16, n3 = N / 64, n4 = N / 256;
  const int c1 = 64, c2 = 128, c3 = 256, c4 = 512;

  // ---- parameter cursor (dict insertion order of make_params) ----
  int cur = 3;                                   // skip p0, x0, o0
  auto nf = [&]() { return (const float*)d_in[cur++]; };
  auto rdTr = [&]() {
    TrW t;
    t.wq = nf(); t.bq = nf(); t.wk = nf(); t.bk = nf(); t.wv = nf(); t.bv = nf();
    t.p1w = nf(); t.p1b = nf(); t.p2w = nf(); t.p2b = nf();
    t.w1w = nf(); t.w1b = nf(); t.w2w = nf(); t.w2b = nf();
    return t;
  };
  auto rdBlk = [&]() { BlkW b; b.l1 = nf(); b.tr = rdTr(); b.l3 = nf(); return b; };
  auto rdEnc = [&]() { EncW e; e.td = nf(); e.b1 = rdBlk(); e.b2 = rdBlk(); return e; };
  auto rdDec = [&]() { DecW d; d.l1w = nf(); d.l1b = nf(); d.l2w = nf(); d.l2b = nf(); return d; };

  EncW enc2 = rdEnc(), enc3 = rdEnc(), enc4 = rdEnc(), enc5 = rdEnc();
  DecW dec5 = rdDec(), dec4 = rdDec(), dec3 = rdDec(), dec2 = rdDec();
  DecW dec1; dec1.l1w = nullptr; dec1.l1b = nullptr; dec1.l2w = nf(); dec1.l2b = nf();
  const float* c1w = nf(); const float* c1b = nf();
  const float* c2w = nf(); const float* c2b = nf();

  // ---- persistent workspace allocations ----
  Pool perm; perm.base = (char*)d_ws; perm.off = 0;
  float* xf  = perm.f((size_t)N * 6);
  float* p1  = perm.f((size_t)n1 * 3);  float* x1  = perm.f((size_t)n1 * c1);
  float* p2  = perm.f((size_t)n2 * 3);  float* x2  = perm.f((size_t)n2 * c2);
  float* p3  = perm.f((size_t)n3 * 3);  float* x3  = perm.f((size_t)n3 * c3);
  float* p4  = perm.f((size_t)n4 * 3);  float* x4  = perm.f((size_t)n4 * c4);
  float* x4h = perm.f((size_t)n4 * c4);
  float* x3n = perm.f((size_t)n3 * c3);
  float* x2n = perm.f((size_t)n2 * c2);
  float* x1n = perm.f((size_t)n1 * c1);
  float* x0d = perm.f((size_t)N * 32);
  float* mean = perm.f(1024);
  float* rstd = perm.f(1024);
  Pool trans = perm;                   // transient pool, reused per macro-step

  // ---- forward ----
  k_concat_rows<<<cdivu((size_t)N * 6, 256), 256, 0, stream>>>(p0, x0, xf, (size_t)N * 6, 3, 3);

  encoder_run(stream, p0, xf, N,  6,   enc2, c1, p1, x1, trans, mean, rstd);
  encoder_run(stream, p1, x1, n1, c1,  enc3, c2, p2, x2, trans, mean, rstd);
  encoder_run(stream, p2, x2, n2, c2,  enc4, c3, p3, x3, trans, mean, rstd);
  encoder_run(stream, p3, x3, n3, c3,  enc5, c4, p4, x4, trans, mean, rstd);

  tu_head_run(stream, x4, n4, c4, dec5, x4h, trans, mean, rstd);
  tu_run(stream, p3, x3, n3, c3, p4, x4h, n4, c4, dec4, x3n, trans, mean, rstd);
  tu_run(stream, p2, x2, n2, c2, p3, x3n, n3, c3, dec3, x2n, trans, mean, rstd);
  tu_run(stream, p1, x1, n1, c1, p2, x2n, n2, c2, dec2, x1n, trans, mean, rstd);
  tu_run(stream, p0, nullptr, N, 32, p1, x1n, n1, c1, dec1, x0d, trans, mean, rstd);

  // classifier head
  Pool cp = trans;
  float* h0 = cp.f((size_t)N * 32);
  gemm(stream, cp, x0d, c1w, c1b, h0, N, 32, 32, 0);
  float* h1 = cp.f((size_t)N * 32);
  bn_act(stream, h0, h1, (size_t)N, 32, nullptr, mean, rstd);
  float* lg = cp.f((size_t)N * NCLS);
  gemm(stream, cp, h1, c2w, c2b, lg, N, 32, NCLS, 0);

  float* out = (float*)d_out;
  k_log_softmax<<<cdivu((size_t)N, 128), 128, 0, stream>>>(lg, out, N, NCLS);
  hipMemcpyAsync(out + (size_t)N * NCLS, x4h, (size_t)n4 * c4 * sizeof(float),
                 hipMemcpyDeviceToDevice, stream);
}